// RGCN_80410377716243
// MI455X (gfx1250) — compile-verified
//
#include <hip/hip_runtime.h>

#define NN 50000
#define NR 4
#define NE 400000
#define INF 128
#define HIDF 128
#define OUTF 64

typedef float v2f __attribute__((ext_vector_type(2)));
typedef float v8f __attribute__((ext_vector_type(8)));

// ---------------------------------------------------------------------------
// Degree counting: one thread per (relation, edge); float atomics so the
// finalize pass can do rsqrt(clamp(deg,1)) in place.
// ---------------------------------------------------------------------------
__global__ void rgcn_count_deg(const int* __restrict__ src,
                               const int* __restrict__ dst,
                               float* __restrict__ deg_out,
                               float* __restrict__ deg_in) {
    int i = blockIdx.x * blockDim.x + threadIdx.x;
    if (i >= NR * NE) return;
    int r = i / NE;
    unsafeAtomicAdd(&deg_out[(size_t)r * NN + src[i]], 1.0f);
    unsafeAtomicAdd(&deg_in [(size_t)r * NN + dst[i]], 1.0f);
}

__global__ void rgcn_finalize_deg(float* __restrict__ deg, int n) {
    int i = blockIdx.x * blockDim.x + threadIdx.x;
    if (i < n) deg[i] = rsqrtf(fmaxf(deg[i], 1.0f));
}

// out[n][f] = sum_r b[r][f]  (accumulator seed; also clears poisoned memory)
__global__ void rgcn_init_bias(float* __restrict__ out,
                               const float* __restrict__ b, int F) {
    int i = blockIdx.x * blockDim.x + threadIdx.x;
    if (i >= NN * F) return;
    int f = i & (F - 1);
    out[i] = b[f] + b[F + f] + b[2 * F + f] + b[3 * F + f];
}

// ---------------------------------------------------------------------------
// hw = (act(A) * scale_row) @ W  via V_WMMA_F32_16X16X4_F32.
//
// One wave owns a 16-row strip x CT columns (CT/16 accumulator tiles): the A
// fragment is loaded once per k-step and feeds CT/16 back-to-back WMMAs,
// cutting A traffic by CT/16 and giving the XDL pipe a dense accumulate chain.
// Compile-time FIN/FOUT fully unroll the K loop (32 steps) so address math
// folds into global_load immediate offsets.
//
// Fragment layouts (ISA 7.12.2, fp32 16x16x4):
//   A: lane<16 -> K=k,k+1 ; lane>=16 -> K=k+2,k+3  (per-lane contiguous pair)
//   B: v0 = W[ka][col], v1 = W[ka+1][col], ka = k + 2*(lane>>4)
//   C/D: VGPR j -> row 16*tm + j + 8*(lane>>4), col = colbase + (lane&15)
// M = 50000 = 3125*16, FOUT in {128,64}: no partial tiles, EXEC all-ones.
// ---------------------------------------------------------------------------
template <int FIN, int FOUT, int CT, int RELU>
__global__ void rgcn_gemm_wmma(const float* __restrict__ A,
                               const float* __restrict__ scale,
                               const float* __restrict__ W,
                               float* __restrict__ C) {
    constexpr int NT = CT / 16;              // column tiles per wave
    constexpr int NG = FOUT / CT;            // column groups
    constexpr int TASKS = (NN / 16) * NG;

    int wave = threadIdx.x >> 5;
    int lane = threadIdx.x & 31;
    int task = blockIdx.x * (blockDim.x >> 5) + wave;
    if (task >= TASKS) return;               // wave-uniform guard (EXEC stays ~0)

    int tm = task / NG;                      // 16-row strip
    int tg = task - tm * NG;                 // column group
    int half = lane >> 4;
    int l15  = lane & 15;
    int row  = tm * 16 + l15;
    int colb = tg * CT + l15;

    float s = scale[row];
    const float* __restrict__ arow = A + (size_t)row * FIN;
    const float* __restrict__ wcol = W + colb;

    v8f acc[NT];
#pragma unroll
    for (int t = 0; t < NT; ++t) acc[t] = (v8f){};

#pragma unroll
    for (int k = 0; k < FIN; k += 4) {
        const int ka = k + 2 * half;
        float a0 = arow[ka];
        float a1 = arow[ka + 1];
        if (RELU) { a0 = fmaxf(a0, 0.0f); a1 = fmaxf(a1, 0.0f); }
        v2f a; a.x = a0 * s; a.y = a1 * s;
#pragma unroll
        for (int t = 0; t < NT; ++t) {
            v2f b; b.x = wcol[(size_t)ka * FOUT + t * 16];
                   b.y = wcol[(size_t)(ka + 1) * FOUT + t * 16];
            acc[t] = __builtin_amdgcn_wmma_f32_16x16x4_f32(
                false, a, false, b, (short)0, acc[t], false, false);
        }
    }

    int r0 = tm * 16 + 8 * half;
#pragma unroll
    for (int t = 0; t < NT; ++t)
#pragma unroll
        for (int j = 0; j < 8; ++j)
            C[(size_t)(r0 + j) * FOUT + colb + t * 16] = acc[t][j];
}

// ---------------------------------------------------------------------------
// Scatter: out[dst[e]] += hw[src[e]] * rsqrt_deg_in[dst[e]]
// One thread per (edge, 4-float chunk): float4 load + 4 global_atomic_add_f32.
// This is the roofline-dominant phase (~2.5 GB irregular L2/HBM traffic).
// ---------------------------------------------------------------------------
__global__ void rgcn_scatter(const int* __restrict__ src,
                             const int* __restrict__ dst,
                             const float* __restrict__ hw,
                             const float* __restrict__ rin,
                             float* __restrict__ out, int Fout) {
    int chunks = Fout >> 2;
    long tid = (long)blockIdx.x * blockDim.x + threadIdx.x;
    long e = tid / chunks;
    int  c = (int)(tid - e * chunks);
    if (e >= NE) return;
    int s = src[e], d = dst[e];
    float sc = rin[d];
    float4 v = ((const float4*)(hw + (size_t)s * Fout))[c];
    float* o = out + (size_t)d * Fout + (size_t)c * 4;
    unsafeAtomicAdd(o + 0, v.x * sc);
    unsafeAtomicAdd(o + 1, v.y * sc);
    unsafeAtomicAdd(o + 2, v.z * sc);
    unsafeAtomicAdd(o + 3, v.w * sc);
}

// ---------------------------------------------------------------------------
extern "C" void kernel_launch(void* const* d_in, const int* in_sizes, int n_in,
                              void* d_out, int out_size, void* d_ws, size_t ws_size,
                              hipStream_t stream) {
    const float* x   = (const float*)d_in[0];
    const int*   src = (const int*)  d_in[1];   // [NR][NE]
    const int*   dst = (const int*)  d_in[2];   // [NR][NE]
    const float* W1  = (const float*)d_in[3];   // [NR][INF][HIDF]
    const float* b1  = (const float*)d_in[4];   // [NR][HIDF]
    const float* W2  = (const float*)d_in[5];   // [NR][HIDF][OUTF]
    const float* b2  = (const float*)d_in[6];   // [NR][OUTF]
    float* out = (float*)d_out;                 // [NN][OUTF]

    // workspace layout
    float* deg_out = (float*)d_ws;                        // NR*NN
    float* deg_in  = deg_out + (size_t)NR * NN;           // NR*NN
    float* h1      = deg_in  + (size_t)NR * NN;           // NN*HIDF
    float* hw      = h1      + (size_t)NN * HIDF;         // NN*HIDF (reused for layer 2)

    const int TPB = 256;                                  // 8 waves / block

    // --- degrees (recomputed every call: deterministic) ---
    hipMemsetAsync(deg_out, 0, (size_t)2 * NR * NN * sizeof(float), stream);
    rgcn_count_deg<<<(NR * NE + TPB - 1) / TPB, TPB, 0, stream>>>(src, dst, deg_out, deg_in);
    rgcn_finalize_deg<<<(2 * NR * NN + TPB - 1) / TPB, TPB, 0, stream>>>(deg_out, 2 * NR * NN);

    // --- layer 1: h1 = sum_r [ scatter_r( (x*rsqrt_out_r) @ W1_r ) * rsqrt_in_r + b1_r ] ---
    rgcn_init_bias<<<(NN * HIDF + TPB - 1) / TPB, TPB, 0, stream>>>(h1, b1, HIDF);
    {
        const int tasks = (NN / 16) * (HIDF / 64);                 // 6250 wave-tasks
        const int gblk  = (tasks + (TPB / 32) - 1) / (TPB / 32);   // 782 blocks
        const long sthreads = (long)NE * (HIDF / 4);
        const int sblk = (int)((sthreads + TPB - 1) / TPB);
        for (int r = 0; r < NR; ++r) {
            rgcn_gemm_wmma<INF, HIDF, 64, 0><<<gblk, TPB, 0, stream>>>(
                x, deg_out + (size_t)r * NN, W1 + (size_t)r * INF * HIDF, hw);
            rgcn_scatter<<<sblk, TPB, 0, stream>>>(
                src + (size_t)r * NE, dst + (size_t)r * NE,
                hw, deg_in + (size_t)r * NN, h1, HIDF);
        }
    }

    // --- layer 2: out = sum_r [ scatter_r( (relu(h1)*rsqrt_out_r) @ W2_r ) * rsqrt_in_r + b2_r ] ---
    rgcn_init_bias<<<(NN * OUTF + TPB - 1) / TPB, TPB, 0, stream>>>(out, b2, OUTF);
    {
        const int tasks = (NN / 16) * (OUTF / 64);                 // 3125 wave-tasks
        const int gblk  = (tasks + (TPB / 32) - 1) / (TPB / 32);   // 391 blocks
        const long sthreads = (long)NE * (OUTF / 4);
        const int sblk = (int)((sthreads + TPB - 1) / TPB);
        for (int r = 0; r < NR; ++r) {
            rgcn_gemm_wmma<HIDF, OUTF, 64, 1><<<gblk, TPB, 0, stream>>>(
                h1, deg_out + (size_t)r * NN, W2 + (size_t)r * HIDF * OUTF, hw);
            rgcn_scatter<<<sblk, TPB, 0, stream>>>(
                src + (size_t)r * NE, dst + (size_t)r * NE,
                hw, deg_in + (size_t)r * NN, out, OUTF);
        }
    }
}